// Postprocess_62921270887070
// MI455X (gfx1250) — compile-verified
//
#include <hip/hip_runtime.h>
#include <stdint.h>

namespace {
constexpr int   kB         = 32;
constexpr int   kNC        = 80;
constexpr int   kN         = 8400;
constexpr int   kNCH       = 4 + kNC;   // 84 channels
constexpr int   kNV4       = kN / 4;    // 2100 float4 columns
constexpr int   kMaxDet    = 300;
constexpr int   kBS        = 1024;      // 32 wave32 waves per workgroup
constexpr int   kNWaves    = kBS / 32;  // 32
constexpr float kScoreThres = 0.005f;
constexpr float kIouThres   = 0.5f;
constexpr float kNeg        = -1e30f;
} // namespace

// --- CDNA5 async global->LDS copy availability (device pass only) ----------
#if defined(__HIP_DEVICE_COMPILE__) && defined(__has_builtin)
#if __has_builtin(__builtin_amdgcn_global_load_async_to_lds_b128)
#define ATHENA_HAVE_ASYNC_LDS 1
#endif
#if __has_builtin(__builtin_amdgcn_s_wait_asynccnt)
#define ATHENA_HAVE_WAIT_ASYNC 1
#endif
#endif

#ifdef ATHENA_HAVE_ASYNC_LDS
typedef int v4i __attribute__((ext_vector_type(4)));
typedef __attribute__((address_space(1))) v4i gv4i_t;   // global int4
typedef __attribute__((address_space(3))) v4i lv4i_t;   // LDS int4
#endif

// Monotonic-orderable key for IEEE floats (bijective; max-reduction safe).
__device__ __forceinline__ unsigned orderKey(float f) {
  unsigned u = __float_as_uint(f);
  return (u & 0x80000000u) ? ~u : (u | 0x80000000u);
}
__device__ __forceinline__ float deKey(unsigned u) {
  return __uint_as_float((u & 0x80000000u) ? (u ^ 0x80000000u) : ~u);
}

__device__ __forceinline__ unsigned long long shflXorU64(unsigned long long v, int mask) {
  unsigned lo = (unsigned)v;
  unsigned hi = (unsigned)(v >> 32);
  lo = __shfl_xor(lo, mask, 32);
  hi = __shfl_xor(hi, mask, 32);
  return ((unsigned long long)hi << 32) | lo;
}

__global__ __launch_bounds__(kBS, 1) void yolo_nms_kernel(const float* __restrict__ x,
                                                          float* __restrict__ out) {
  // ---- LDS-resident per-image working set (~204 KB; CDNA5 WGP has 320 KB) --
  __shared__ __align__(16) float    s_box[4 * kN];  // raw x1,y1,x2,y2 rows (SoA)
  __shared__ __align__(16) unsigned s_ukey[kN];     // orderable score keys
  __shared__ __align__(16) int      s_cls[kN];      // argmax class per anchor
  __shared__ __align__(8)  unsigned long long s_red[kNWaves];
  __shared__ float s_bc[8];                         // bcast: ob box, area, offset
  __shared__ int   s_seli[kMaxDet];
  __shared__ float s_selsc[kMaxDet];
  __shared__ int   s_mi[2];                         // selected index, ok flag

  const int tid = threadIdx.x;
  const int b   = blockIdx.x;
  const float* __restrict__ xb = x + (size_t)b * kNCH * kN;

  // ---- Stage 1a: kick off async copy of the 4 box rows into LDS -----------
#ifdef ATHENA_HAVE_ASYNC_LDS
  for (int r = 0; r < 4; ++r) {
    const float* grow = xb + r * kN;
    float*       lrow = s_box + r * kN;
    for (int t = tid; t < kNV4; t += kBS) {
      gv4i_t* gp = (gv4i_t*)(uintptr_t)(grow + 4 * t);
      lv4i_t* lp = (lv4i_t*)(uint32_t)(uintptr_t)(lrow + 4 * t);
      __builtin_amdgcn_global_load_async_to_lds_b128(gp, lp, 0, 0);
    }
  }
#else
  for (int r = 0; r < 4; ++r) {
    const float* grow = xb + r * kN;
    float*       lrow = s_box + r * kN;
    for (int t = tid; t < kNV4; t += kBS)
      *(float4*)(lrow + 4 * t) = *(const float4*)(grow + 4 * t);
  }
#endif

  // ---- Stage 1b: stream 80 class rows; per-anchor max + first-argmax ------
  for (int t = tid; t < kNV4; t += kBS) {
    const float* colBase = xb + 4 * kN + 4 * t;
    float4 bv = *(const float4*)colBase;
    int cx = 0, cy = 0, cz = 0, cw = 0;
    for (int c = 1; c < kNC; ++c) {
      if ((c & 7) == 1) __builtin_prefetch(colBase + (c + 8) * kN, 0, 0);
      float4 v = *(const float4*)(colBase + c * kN);
      if (v.x > bv.x) { bv.x = v.x; cx = c; }
      if (v.y > bv.y) { bv.y = v.y; cy = c; }
      if (v.z > bv.z) { bv.z = v.z; cz = c; }
      if (v.w > bv.w) { bv.w = v.w; cw = c; }
    }
    uint4 uk;
    uk.x = orderKey(bv.x > kScoreThres ? bv.x : kNeg);
    uk.y = orderKey(bv.y > kScoreThres ? bv.y : kNeg);
    uk.z = orderKey(bv.z > kScoreThres ? bv.z : kNeg);
    uk.w = orderKey(bv.w > kScoreThres ? bv.w : kNeg);
    *(uint4*)(s_ukey + 4 * t) = uk;
    int4 ci; ci.x = cx; ci.y = cy; ci.z = cz; ci.w = cw;
    *(int4*)(s_cls + 4 * t) = ci;
  }

#ifdef ATHENA_HAVE_ASYNC_LDS
#ifdef ATHENA_HAVE_WAIT_ASYNC
  __builtin_amdgcn_s_wait_asynccnt(0);
#else
  asm volatile("s_wait_asynccnt 0" ::: "memory");
#endif
#endif
  __syncthreads();

  // ---- Stage 2: offset = max(all box coords) + 1 --------------------------
  float mx = -3.4e38f;
  for (int t = tid; t < kN; t += kBS) {
    float4 v0 = *(const float4*)(s_box + 0 * kN + 0);  // placeholder removed below
    (void)v0; break;
  }
  mx = -3.4e38f;
  for (int t = tid; t < 4 * kN; t += kBS) mx = fmaxf(mx, s_box[t]);
  for (int o = 16; o > 0; o >>= 1) mx = fmaxf(mx, __shfl_xor(mx, o, 32));
  float* s_redf = (float*)s_red;
  if ((tid & 31) == 0) s_redf[tid >> 5] = mx;
  __syncthreads();
  if (tid < 32) {
    float m = s_redf[tid];
    for (int o = 16; o > 0; o >>= 1) m = fmaxf(m, __shfl_xor(m, o, 32));
    if (tid == 0) s_bc[6] = m + 1.0f;
  }
  __syncthreads();
  const float offset = s_bc[6];
  const unsigned kNegKey = orderKey(kNeg);

  // ---- Stage 3: greedy class-aware NMS, fully LDS-resident ----------------
  int nsel = kMaxDet;
  for (int it = 0; it < kMaxDet; ++it) {
    // global argmax via packed (ukey || (N-1-j)) max-reduction, float4-wide scan
    unsigned long long key = 0ull;
    for (int t = tid; t < kNV4; t += kBS) {
      uint4 u = *(const uint4*)(s_ukey + 4 * t);
      int j0 = 4 * t;
      unsigned long long k0 = ((unsigned long long)u.x << 32) | (unsigned)(kN - 1 - (j0 + 0));
      unsigned long long k1 = ((unsigned long long)u.y << 32) | (unsigned)(kN - 1 - (j0 + 1));
      unsigned long long k2 = ((unsigned long long)u.z << 32) | (unsigned)(kN - 1 - (j0 + 2));
      unsigned long long k3 = ((unsigned long long)u.w << 32) | (unsigned)(kN - 1 - (j0 + 3));
      k0 = (k1 > k0) ? k1 : k0;
      k2 = (k3 > k2) ? k3 : k2;
      k0 = (k2 > k0) ? k2 : k0;
      key = (k0 > key) ? k0 : key;
    }
    for (int o = 16; o > 0; o >>= 1) {
      unsigned long long k2 = shflXorU64(key, o);
      key = (k2 > key) ? k2 : key;
    }
    if ((tid & 31) == 0) s_red[tid >> 5] = key;
    __syncthreads();
    if (tid < 32) {
      unsigned long long k = s_red[tid];
      for (int o = 16; o > 0; o >>= 1) {
        unsigned long long k2 = shflXorU64(k, o);
        k = (k2 > k) ? k2 : k;
      }
      if (tid == 0) {
        int   i  = kN - 1 - (int)(k & 0xFFFFFFFFull);
        float sv = deKey((unsigned)(k >> 32));
        int   ok = sv > (0.5f * kNeg);
        float co = (float)s_cls[i] * offset;   // class-offset box (matches ref ob)
        float ox1 = s_box[0 * kN + i] + co;
        float oy1 = s_box[1 * kN + i] + co;
        float ox2 = s_box[2 * kN + i] + co;
        float oy2 = s_box[3 * kN + i] + co;
        s_bc[0] = ox1; s_bc[1] = oy1; s_bc[2] = ox2; s_bc[3] = oy2;
        s_bc[4] = (ox2 - ox1) * (oy2 - oy1);
        s_mi[0] = i;
        s_mi[1] = ok;
        s_seli[it]  = i;
        s_selsc[it] = sv;
      }
    }
    __syncthreads();
    if (!s_mi[1]) { nsel = it; break; }   // uniform: once exhausted, stays exhausted
    const int   isel = s_mi[0];
    const float bx1 = s_bc[0], by1 = s_bc[1], bx2 = s_bc[2], by2 = s_bc[3];
    const float ai  = s_bc[4];
    for (int t = tid; t < kNV4; t += kBS) {
      const int j0 = 4 * t;
      float4 vx1 = *(const float4*)(s_box + 0 * kN + j0);
      float4 vy1 = *(const float4*)(s_box + 1 * kN + j0);
      float4 vx2 = *(const float4*)(s_box + 2 * kN + j0);
      float4 vy2 = *(const float4*)(s_box + 3 * kN + j0);
      int4   vc  = *(const int4*)(s_cls + j0);
      uint4  vu  = *(const uint4*)(s_ukey + j0);
#define ATHENA_SUPPRESS(comp, jj)                                            \
      {                                                                      \
        float co = (float)vc.comp * offset;                                  \
        float x1 = vx1.comp + co, y1 = vy1.comp + co;                        \
        float x2 = vx2.comp + co, y2 = vy2.comp + co;                        \
        float w = fmaxf(fminf(bx2, x2) - fmaxf(bx1, x1), 0.0f);              \
        float h = fmaxf(fminf(by2, y2) - fmaxf(by1, y1), 0.0f);              \
        float inter = w * h;                                                 \
        float aj = (x2 - x1) * (y2 - y1);                                    \
        float iou = inter / (ai + aj - inter + 1e-9f);                       \
        if (iou > kIouThres || (jj) == isel) vu.comp = kNegKey;              \
      }
      ATHENA_SUPPRESS(x, j0 + 0)
      ATHENA_SUPPRESS(y, j0 + 1)
      ATHENA_SUPPRESS(z, j0 + 2)
      ATHENA_SUPPRESS(w, j0 + 3)
#undef ATHENA_SUPPRESS
      *(uint4*)(s_ukey + j0) = vu;   // lane-exclusive slots: race-free
    }
    __syncthreads();
  }

  // ---- Stage 4: emit dets [300,6] (raw boxes, exact) + count --------------
  float* outb = out + (size_t)b * kMaxDet * 6;
  for (int t = tid; t < kMaxDet; t += kBS) {
    float r0 = 0.f, r1 = 0.f, r2 = 0.f, r3 = 0.f, r4 = 0.f, r5 = 0.f;
    if (t < nsel) {
      int i = s_seli[t];
      r0 = s_box[0 * kN + i];
      r1 = s_box[1 * kN + i];
      r2 = s_box[2 * kN + i];
      r3 = s_box[3 * kN + i];
      r4 = s_selsc[t];
      r5 = (float)s_cls[i];
    }
    float* p = outb + t * 6;
    p[0] = r0; p[1] = r1; p[2] = r2; p[3] = r3; p[4] = r4; p[5] = r5;
  }
  if (tid == 0) out[(size_t)kB * kMaxDet * 6 + b] = (float)nsel;
}

extern "C" void kernel_launch(void* const* d_in, const int* in_sizes, int n_in,
                              void* d_out, int out_size, void* d_ws, size_t ws_size,
                              hipStream_t stream) {
  (void)in_sizes; (void)n_in; (void)d_ws; (void)ws_size; (void)out_size;
  const float* x  = (const float*)d_in[0];   // [32, 84, 8400] float32
  float* out      = (float*)d_out;           // 32*300*6 dets + 32 counts
  hipLaunchKernelGGL(yolo_nms_kernel, dim3(kB), dim3(kBS), 0, stream, x, out);
}